// GraphSAGE_5626407158206
// MI455X (gfx1250) — compile-verified
//
#include <hip/hip_runtime.h>

#define NN 50000
#define NE 1600000
#define DD 128
#define NG 128

typedef float v2f __attribute__((ext_vector_type(2)));
typedef float v8f __attribute__((ext_vector_type(8)));

__global__ __launch_bounds__(256) void zero_f32(float* __restrict__ p, int n4) {
    int i = blockIdx.x * 256 + threadIdx.x;
    if (i < n4) ((float4*)p)[i] = make_float4(0.f, 0.f, 0.f, 0.f);
}

__global__ __launch_bounds__(256) void degree_kernel(const int* __restrict__ dst,
                                                     float* __restrict__ deg, int nE) {
    int e = blockIdx.x * 256 + threadIdx.x;
    if (e < nE) atomicAdd(&deg[dst[e]], 1.0f);
}

__global__ __launch_bounds__(256) void invdeg_kernel(float* __restrict__ deg, int n) {
    int i = blockIdx.x * 256 + threadIdx.x;
    if (i < n) {
        float d = deg[i];
        deg[i] = d > 0.f ? 1.f / d : 0.f;
    }
}

// One wave32 per edge: lane L moves 4 consecutive floats (float4 load + 4 f32 atomics).
__global__ __launch_bounds__(256) void scatter_kernel(const float* __restrict__ x,
                                                      const int* __restrict__ src,
                                                      const int* __restrict__ dst,
                                                      float* __restrict__ agg, int nE) {
    int lane = threadIdx.x & 31;
    int e = (blockIdx.x * 256 + threadIdx.x) >> 5;
    if (e >= nE) return;
    int s = src[e];
    int d = dst[e];
    const float4 v = *(const float4*)(x + (size_t)s * DD + lane * 4);
    float* p = agg + (size_t)d * DD + lane * 4;
    atomicAdd(p + 0, v.x);
    atomicAdd(p + 1, v.y);
    atomicAdd(p + 2, v.z);
    atomicAdd(p + 3, v.w);
}

// out[m,n] = (inv[m]*agg[m,:]) @ Wl[:,n] + x[m,:] @ Wr[:,n] + b[n], optional ReLU.
// One wave per 16x16 tile; block of 8 waves covers all 128 columns; grid.x = 3125 M-tiles.
// V_WMMA_F32_16X16X4_F32 fragment layout:
//   A (16x4): lane m(0-15)/m+16 holds K = 2*hi + {0,1} in VGPR {0,1}  -> float2 load
//   B (4x16): lane holds column n0+(lane&15); VGPR {0,1} = rows 2*hi + {0,1}
//   C/D: lane holds column n0+(lane&15); VGPR v = row v + 8*hi
__global__ __launch_bounds__(256) void sage_gemm(const float* __restrict__ agg,
                                                 const float* __restrict__ inv,
                                                 const float* __restrict__ xin,
                                                 const float* __restrict__ Wl,
                                                 const float* __restrict__ Wr,
                                                 const float* __restrict__ bias,
                                                 float* __restrict__ out, int relu) {
    const int lane = threadIdx.x & 31;
    const int wave = threadIdx.x >> 5;
    const int hi   = lane >> 4;       // which K/M half this lane covers
    const int mr   = lane & 15;       // A row / B-C column within tile
    const int m0   = blockIdx.x * 16;
    const int n0   = wave * 16;
    const int kb   = 2 * hi;

    const float scale = inv[m0 + mr];
    const float* a1 = agg + (size_t)(m0 + mr) * DD;
    const float* a2 = xin + (size_t)(m0 + mr) * DD;

    v8f c = {0.f, 0.f, 0.f, 0.f, 0.f, 0.f, 0.f, 0.f};

    #pragma unroll 8
    for (int k = 0; k < DD; k += 4) {
        v2f a = *(const v2f*)(a1 + k + kb);
        a.x *= scale;
        a.y *= scale;
        v2f bf;
        bf.x = Wl[(size_t)(k + kb + 0) * DD + n0 + mr];
        bf.y = Wl[(size_t)(k + kb + 1) * DD + n0 + mr];
        c = __builtin_amdgcn_wmma_f32_16x16x4_f32(false, a, false, bf, (short)0, c,
                                                  false, false);
    }
    #pragma unroll 8
    for (int k = 0; k < DD; k += 4) {
        v2f a = *(const v2f*)(a2 + k + kb);
        v2f bf;
        bf.x = Wr[(size_t)(k + kb + 0) * DD + n0 + mr];
        bf.y = Wr[(size_t)(k + kb + 1) * DD + n0 + mr];
        c = __builtin_amdgcn_wmma_f32_16x16x4_f32(false, a, false, bf, (short)0, c,
                                                  false, false);
    }

    const float bv = bias[n0 + mr];
    #pragma unroll
    for (int v = 0; v < 8; ++v) {
        float val = c[v] + bv;
        if (relu) val = fmaxf(val, 0.f);
        out[(size_t)(m0 + v + 8 * hi) * DD + n0 + mr] = val;
    }
}

// One wave32 per node: global_add_pool via f32 atomics into [128 x 128] output.
__global__ __launch_bounds__(256) void pool_kernel(const float* __restrict__ x,
                                                   const int* __restrict__ batch,
                                                   float* __restrict__ out, int n) {
    int lane = threadIdx.x & 31;
    int node = (blockIdx.x * 256 + threadIdx.x) >> 5;
    if (node >= n) return;
    int g = batch[node];
    const float4 v = *(const float4*)(x + (size_t)node * DD + lane * 4);
    float* p = out + (size_t)g * DD + lane * 4;
    atomicAdd(p + 0, v.x);
    atomicAdd(p + 1, v.y);
    atomicAdd(p + 2, v.z);
    atomicAdd(p + 3, v.w);
}

extern "C" void kernel_launch(void* const* d_in, const int* in_sizes, int n_in,
                              void* d_out, int out_size, void* d_ws, size_t ws_size,
                              hipStream_t stream) {
    const float* x_in = (const float*)d_in[0];
    const int*   eidx = (const int*)d_in[1];   // [2, NE]: src then dst
    const int*   batch = (const int*)d_in[2];  // [NN]
    const float* W_l  = (const float*)d_in[3]; // [3,128,128]
    const float* W_r  = (const float*)d_in[4]; // [3,128,128]
    const float* bias = (const float*)d_in[5]; // [3,128]
    const int* src = eidx;
    const int* dst = eidx + NE;

    const size_t ND = (size_t)NN * DD;
    float* xa  = (float*)d_ws;
    float* xb  = xa + ND;
    float* agg = xb + ND;
    float* inv = agg + ND;

    // Degree (once; layer-invariant), then invert in place.
    zero_f32<<<(NN / 4 + 255) / 256, 256, 0, stream>>>(inv, NN / 4);
    degree_kernel<<<NE / 256, 256, 0, stream>>>(dst, inv, NE);
    invdeg_kernel<<<(NN + 255) / 256, 256, 0, stream>>>(inv, NN);

    // Working copy of node features (inputs must stay unmodified).
    hipMemcpyAsync(xa, x_in, ND * sizeof(float), hipMemcpyDeviceToDevice, stream);

    float* cur = xa;
    float* nxt = xb;
    for (int l = 0; l < 3; ++l) {
        zero_f32<<<(int)(ND / 4) / 256, 256, 0, stream>>>(agg, (int)(ND / 4));
        scatter_kernel<<<(NE * 32) / 256, 256, 0, stream>>>(cur, src, dst, agg, NE);
        sage_gemm<<<NN / 16, 256, 0, stream>>>(agg, inv, cur,
                                               W_l + (size_t)l * DD * DD,
                                               W_r + (size_t)l * DD * DD,
                                               bias + (size_t)l * DD,
                                               nxt, (l < 2) ? 1 : 0);
        float* t = cur; cur = nxt; nxt = t;
    }

    zero_f32<<<(NG * DD / 4 + 255) / 256, 256, 0, stream>>>((float*)d_out, NG * DD / 4);
    pool_kernel<<<(NN * 32) / 256, 256, 0, stream>>>(cur, batch, (float*)d_out, NN);
}